// TrajectoryFK_88510686036542
// MI455X (gfx1250) — compile-verified
//
#include <hip/hip_runtime.h>
#include <hip/hip_bf16.h>

// ---------------- problem constants ----------------
#define LSTEPS 65536
#define NJ     22
#define ROWF   (NJ * 6)      // 132 floats per timestep row
#define CHUNK  64
#define NCH    (LSTEPS / CHUNK)   // 1024 chunks

static constexpr int PAR[NJ] = {-1, 0, 0, 0, 1, 2, 3, 4, 5, 6, 7, 8,
                                 9, 9, 9, 12, 13, 14, 16, 17, 18, 19};

typedef __attribute__((ext_vector_type(2))) float v2f;
typedef __attribute__((ext_vector_type(8))) float v8f;

// ---------------- small mat helpers ----------------
__device__ __forceinline__ void gs6(const float* __restrict__ d, float* __restrict__ M) {
    // 6D continuous rotation -> 3x3 (columns b1,b2,b3), row-major M[r*3+c]
    float a1x = d[0], a1y = d[1], a1z = d[2];
    float a2x = d[3], a2y = d[4], a2z = d[5];
    float n1 = sqrtf(fmaf(a1x, a1x, fmaf(a1y, a1y, a1z * a1z)));
    float i1 = 1.0f / fmaxf(n1, 1e-12f);
    float b1x = a1x * i1, b1y = a1y * i1, b1z = a1z * i1;
    float dp = fmaf(b1x, a2x, fmaf(b1y, a2y, b1z * a2z));
    float b2x = a2x - b1x * dp, b2y = a2y - b1y * dp, b2z = a2z - b1z * dp;
    float n2 = sqrtf(fmaf(b2x, b2x, fmaf(b2y, b2y, b2z * b2z)));
    float i2 = 1.0f / fmaxf(n2, 1e-12f);
    b2x *= i2; b2y *= i2; b2z *= i2;
    float b3x = fmaf(b1y, b2z, -b1z * b2y);
    float b3y = fmaf(b1z, b2x, -b1x * b2z);
    float b3z = fmaf(b1x, b2y, -b1y * b2x);
    M[0] = b1x; M[1] = b2x; M[2] = b3x;
    M[3] = b1y; M[4] = b2y; M[5] = b3y;
    M[6] = b1z; M[7] = b2z; M[8] = b3z;
}

__device__ __forceinline__ void mm3(const float* __restrict__ A,
                                    const float* __restrict__ B,
                                    float* __restrict__ C) {
#pragma unroll
    for (int r = 0; r < 3; ++r)
#pragma unroll
        for (int c = 0; c < 3; ++c)
            C[r * 3 + c] = fmaf(A[r * 3 + 0], B[0 + c],
                           fmaf(A[r * 3 + 1], B[3 + c],
                                A[r * 3 + 2] * B[6 + c]));
}

__device__ __forceinline__ void mv3(const float* __restrict__ A,
                                    const float* __restrict__ x,
                                    float* __restrict__ y) {
#pragma unroll
    for (int r = 0; r < 3; ++r)
        y[r] = fmaf(A[r * 3 + 0], x[0], fmaf(A[r * 3 + 1], x[1], A[r * 3 + 2] * x[2]));
}

// ================= kernel 1: per-chunk delta-matrix aggregate =================
__global__ void k_agg(const float* __restrict__ pred, float* __restrict__ agg) {
    int c = blockIdx.x * blockDim.x + threadIdx.x;   // 0..NCH-1
    const float* base = pred + (size_t)c * CHUNK * ROWF;
    float R[9] = {1, 0, 0, 0, 1, 0, 0, 0, 1};
    for (int s = 0; s < CHUNK; ++s) {
        const float* row = base + s * ROWF;
        __builtin_prefetch(row + 8 * ROWF, 0, 0);
        float M[9], T[9];
        gs6(row + 6, M);                 // joint 1 delta
        mm3(R, M, T);
#pragma unroll
        for (int q = 0; q < 9; ++q) R[q] = T[q];
    }
#pragma unroll
    for (int q = 0; q < 9; ++q) agg[c * 9 + q] = R[q];
}

// ============== kernel 2: Hillis-Steele scan of 1024 matrices (1 block) ==============
__global__ void k_scan_mat(const float* __restrict__ agg, float* __restrict__ pfx) {
    __shared__ float s[NCH * 9];
    int i = threadIdx.x;
    float r[9];
#pragma unroll
    for (int q = 0; q < 9; ++q) { r[q] = agg[i * 9 + q]; s[i * 9 + q] = r[q]; }
    __syncthreads();
    for (int off = 1; off < NCH; off <<= 1) {
        float p[9];
        bool has = (i >= off);
        if (has) {
#pragma unroll
            for (int q = 0; q < 9; ++q) p[q] = s[(i - off) * 9 + q];
        }
        __syncthreads();
        if (has) {
            float t[9];
            mm3(p, r, t);                 // earlier chunk on the left
#pragma unroll
            for (int q = 0; q < 9; ++q) { r[q] = t[q]; s[i * 9 + q] = t[q]; }
        }
        __syncthreads();
    }
    // exclusive prefix: P_0 = I, P_i = S_{i-1}
    if (i == 0) {
        const float I9[9] = {1, 0, 0, 0, 1, 0, 0, 0, 1};
#pragma unroll
        for (int q = 0; q < 9; ++q) pfx[q] = I9[q];
    } else {
#pragma unroll
        for (int q = 0; q < 9; ++q) pfx[i * 9 + q] = s[(i - 1) * 9 + q];
    }
}

// ===== kernel 3: apply carries; emit R_t, within-chunk vel cumsum, chunk vel sums =====
__global__ void k_rvel(const float* __restrict__ pred, const float* __restrict__ pfx,
                       float* __restrict__ Rw, float* __restrict__ cum,
                       float* __restrict__ vsum) {
    int c = blockIdx.x * blockDim.x + threadIdx.x;
    const float* base = pred + (size_t)c * CHUNK * ROWF;
    float R[9];
#pragma unroll
    for (int q = 0; q < 9; ++q) R[q] = pfx[c * 9 + q];
    float acc[3] = {0.f, 0.f, 0.f};
    for (int s = 0; s < CHUNK; ++s) {
        int t = c * CHUNK + s;
        const float* row = base + s * ROWF;
        __builtin_prefetch(row + 8 * ROWF, 0, 0);
        float M[9], T[9];
        gs6(row + 6, M);
        mm3(R, M, T);
#pragma unroll
        for (int q = 0; q < 9; ++q) R[q] = T[q];
#pragma unroll
        for (int q = 0; q < 9; ++q) Rw[(size_t)t * 9 + q] = R[q];
        float vel[3] = {row[0], row[1], row[2]};
        float gv[3];
        mv3(R, vel, gv);
#pragma unroll
        for (int q = 0; q < 3; ++q) { acc[q] += gv[q]; cum[t * 3 + q] = acc[q]; }
    }
#pragma unroll
    for (int q = 0; q < 3; ++q) vsum[c * 3 + q] = acc[q];
}

// ===== kernel 4: vec3 prefix-sum over 1024 chunk sums via WMMA lower-triangular matmul =====
// D(16x16) = Ltri(16x16) * V(16x3-padded) built from 4 chained V_WMMA_F32_16X16X4_F32.
__global__ void k_vscan(const float* __restrict__ vsum, float* __restrict__ vcar) {
    __shared__ float s_incl[64][16][3];
    __shared__ float s_carry[64][3];
    const int tid  = threadIdx.x;
    const int lane = tid & 31;
    const int wave = tid >> 5;
    const int m  = lane & 15;          // row (A/M) or column (B/D) index
    const int hi = lane >> 4;          // lane group: 0 -> K=0/1, 1 -> K=2/3

    for (int k = wave; k < 64; k += 8) {          // 8 waves x 8 tiles; uniform per wave
        v8f cacc = {};
#pragma unroll
        for (int j = 0; j < 4; ++j) {
            const int kx = 4 * j + (hi ? 2 : 0);  // K index of vgpr .x for this lane
            const int ky = kx + 1;                // K index of vgpr .y
            v2f a, b;
            a.x = (kx <= m) ? 1.0f : 0.0f;        // inclusive lower-triangular ones
            a.y = (ky <= m) ? 1.0f : 0.0f;
            float bx = 0.0f, by = 0.0f;
            if (m < 3) {                          // only 3 columns of B are real data
                bx = vsum[(k * 16 + kx) * 3 + m];
                by = vsum[(k * 16 + ky) * 3 + m];
            }
            b.x = bx; b.y = by;
            cacc = __builtin_amdgcn_wmma_f32_16x16x4_f32(
                false, a, false, b, (short)0, cacc, false, false);
        }
        if (m < 3) {
#pragma unroll
            for (int r = 0; r < 8; ++r)           // C/D: vgpr r -> row r (lanes<16) / r+8
                s_incl[k][r + hi * 8][m] = cacc[r];
        }
    }
    __syncthreads();
    if (tid == 0) {                                // serial scan of 64 tile totals
        s_carry[0][0] = s_carry[0][1] = s_carry[0][2] = 0.f;
        for (int k = 1; k < 64; ++k)
            for (int q = 0; q < 3; ++q)
                s_carry[k][q] = s_carry[k - 1][q] + s_incl[k - 1][15][q];
    }
    __syncthreads();
    for (int idx = tid; idx < NCH; idx += blockDim.x) {
        int k = idx >> 4, mm = idx & 15;
#pragma unroll
        for (int q = 0; q < 3; ++q)                // exclusive = carry + inclusive - own
            vcar[idx * 3 + q] = s_carry[k][q] + s_incl[k][mm][q] - vsum[idx * 3 + q];
    }
}

// ================= kernel 5: per-timestep forward kinematics =================
__global__ void k_fk(const float* __restrict__ pred, const float* __restrict__ offs,
                     const float* __restrict__ Rw, const float* __restrict__ cum,
                     const float* __restrict__ vcar, float* __restrict__ out) {
    int t = blockIdx.x * blockDim.x + threadIdx.x;
    int c = t >> 6;  // CHUNK = 64
    float R[9];
#pragma unroll
    for (int q = 0; q < 9; ++q) R[q] = Rw[(size_t)t * 9 + q];

    float rots[NJ][9];
    float pos[NJ][3];
#pragma unroll
    for (int q = 0; q < 9; ++q) { rots[0][q] = R[q]; rots[1][q] = R[q]; }
#pragma unroll
    for (int q = 0; q < 3; ++q) pos[0][q] = vcar[c * 3 + q] + cum[t * 3 + q];

    float* ob = out + (size_t)t * NJ * 3;
#pragma unroll
    for (int q = 0; q < 3; ++q) ob[q] = pos[0][q];

    {   // joint 1: same rotation as root, offset displacement
        float o1[3] = {offs[3], offs[4], offs[5]};
        float d1[3];
        mv3(R, o1, d1);
#pragma unroll
        for (int q = 0; q < 3; ++q) { pos[1][q] = pos[0][q] + d1[q]; ob[3 + q] = pos[1][q]; }
    }

    const float* row = pred + (size_t)t * ROWF;
#pragma unroll
    for (int i = 2; i < NJ; ++i) {
        const int p = PAR[i];
        float M[9];
        gs6(row + i * 6, M);
        mm3(rots[p], M, rots[i]);
        float o[3] = {offs[i * 3 + 0], offs[i * 3 + 1], offs[i * 3 + 2]};
        float dv[3];
        mv3(rots[p], o, dv);
#pragma unroll
        for (int q = 0; q < 3; ++q) {
            pos[i][q] = pos[p][q] + dv[q];
            ob[i * 3 + q] = pos[i][q];
        }
    }
}

// ---------------- launcher ----------------
extern "C" void kernel_launch(void* const* d_in, const int* in_sizes, int n_in,
                              void* d_out, int out_size, void* d_ws, size_t ws_size,
                              hipStream_t stream) {
    const float* pred = (const float*)d_in[0];   // (65536, 22, 6) f32
    const float* offs = (const float*)d_in[1];   // (22, 3) f32
    float* out = (float*)d_out;                  // (65536, 22, 3) f32
    float* ws  = (float*)d_ws;

    // workspace layout (floats)
    float* agg  = ws;                               // NCH*9      =   9216
    float* pfx  = agg  + NCH * 9;                   // NCH*9      =   9216
    float* Rw   = pfx  + NCH * 9;                   // L*9        = 589824
    float* cum  = Rw   + (size_t)LSTEPS * 9;        // L*3        = 196608
    float* vsum = cum  + (size_t)LSTEPS * 3;        // NCH*3      =   3072
    float* vcar = vsum + NCH * 3;                   // NCH*3      =   3072
    (void)in_sizes; (void)n_in; (void)out_size; (void)ws_size;

    k_agg     <<<NCH / 256, 256, 0, stream>>>(pred, agg);
    k_scan_mat<<<1, NCH, 0, stream>>>(agg, pfx);
    k_rvel    <<<NCH / 256, 256, 0, stream>>>(pred, pfx, Rw, cum, vsum);
    k_vscan   <<<1, 256, 0, stream>>>(vsum, vcar);
    k_fk      <<<LSTEPS / 256, 256, 0, stream>>>(pred, offs, Rw, cum, vcar, out);
}